// Attention_13288628814607
// MI455X (gfx1250) — compile-verified
//
#include <hip/hip_runtime.h>
#include <hip/hip_bf16.h>

// ---- problem constants (match reference) ----
#define SEQ    1024
#define CH     64
#define UNITS  32
#define WIN    64
#define QT     16          // queries per workgroup tile
#define KEYS   (QT + WIN)  // 80 banded key rows per tile
#define BATCH  4
#define EPSF   1e-7f

typedef float v2f __attribute__((ext_vector_type(2)));
typedef float v8f __attribute__((ext_vector_type(8)));

// D = A(16xK) * B(Kx16) accumulated with V_WMMA_F32_16X16X4_F32.
// A: row-major LDS, leading dim LDA.  B element (k,n) at B[k*BK + n*BN].
// f32 WMMA register layout (ISA 7.12.2):
//   A: lanes 0-15 -> row M=lane, v0=K(k0), v1=K(k0+1); lanes 16-31 -> same rows, K(k0+2),K(k0+3)
//   B: mirror over N;  C/D: N = lane%16, M = vgpr + 8*(lane/16)
template <int K, int LDA, int BK, int BN>
__device__ __forceinline__ v8f wmma_f32_tile(const float* __restrict__ A,
                                             const float* __restrict__ B,
                                             int lane) {
  v8f acc = {0.f, 0.f, 0.f, 0.f, 0.f, 0.f, 0.f, 0.f};
  const int l  = lane & 15;
  const int ko = (lane >> 4) << 1;   // 0 for lanes 0-15, 2 for lanes 16-31
#pragma unroll
  for (int k0 = 0; k0 < K; k0 += 4) {
    v2f av, bv;
    av.x = A[l * LDA + (k0 + ko)];
    av.y = A[l * LDA + (k0 + ko + 1)];
    bv.x = B[(k0 + ko) * BK + l * BN];
    bv.y = B[(k0 + ko + 1) * BK + l * BN];
    acc = __builtin_amdgcn_wmma_f32_16x16x4_f32(
        /*neg_a=*/false, av, /*neg_b=*/false, bv,
        /*c_mod=*/(short)0, acc, /*reuse_a=*/false, /*reuse_b=*/false);
  }
  return acc;
}

__device__ __forceinline__ void store_tile_lds(float* dst, int ld, v8f d, int lane) {
  const int l  = lane & 15;
  const int mo = (lane >> 4) * 8;
#pragma unroll
  for (int j = 0; j < 8; ++j) dst[(mo + j) * ld + l] = d[j];
}

__launch_bounds__(32)
__global__ void Attention_banded_wmma(const float* __restrict__ x,     // (B, C, S)
                                      const float* __restrict__ Wt,    // (U, C)
                                      const float* __restrict__ Wx,    // (U, C)
                                      const float* __restrict__ Wa_w,  // (1, U)
                                      const float* __restrict__ Wa_b,  // (1,)
                                      const float* __restrict__ bh,    // (U,)
                                      float* __restrict__ out_v,       // (B, C, S)
                                      float* __restrict__ out_a) {     // (B, S, S)
  __shared__ float sXq[QT][CH];     // query rows (row-major, xt layout)
  __shared__ float sXk[KEYS][CH];   // banded key rows, zero-padded OOB
  __shared__ float sWt[UNITS][CH];
  __shared__ float sWx[UNITS][CH];
  __shared__ float sQ[QT][UNITS];
  __shared__ float sK[KEYS][UNITS];
  __shared__ float sA[QT][KEYS];    // banded attention weights
  __shared__ float sWa[UNITS];
  __shared__ float sBh[UNITS];

  const int lane = threadIdx.x;          // wave32: 0..31
  const int s0   = blockIdx.x * QT;      // first query of tile
  const int b    = blockIdx.y;
  const int t0   = s0 - WIN / 2;         // first banded key index (may be <0)

  const float* xb = x + (size_t)b * CH * SEQ;
  __builtin_prefetch(xb, 0, 3);          // global_prefetch_b8

  // ---- stage weights + biases ----
  for (int i = lane; i < UNITS * CH; i += 32) {
    sWt[0][i] = Wt[i];
    sWx[0][i] = Wx[i];
  }
  sWa[lane] = Wa_w[lane];
  sBh[lane] = bh[lane];

  // ---- stage xt query rows: xt[s][c] = x[b][c][s] ----
  for (int i = lane; i < QT * CH; i += 32) {
    const int r = i >> 6, c = i & 63;
    sXq[r][c] = xb[c * SEQ + (s0 + r)];
  }
  // ---- stage banded key rows (zero-pad out of range) ----
  for (int i = lane; i < KEYS * CH; i += 32) {
    const int r = i >> 6, c = i & 63;
    const int t = t0 + r;
    sXk[r][c] = (t >= 0 && t < SEQ) ? xb[c * SEQ + t] : 0.0f;
  }
  __syncthreads();

  // zero the banded A tile (band writes overwrite live entries later)
  for (int i = lane; i < QT * KEYS; i += 32) sA[0][i] = 0.0f;

  // ---- Q = Xq(16x64) * Wt^T(64x32): B(k,n) = Wt[n][k] -> BK=1, BN=CH ----
#pragma unroll
  for (int n0 = 0; n0 < UNITS; n0 += 16) {
    v8f d = wmma_f32_tile<CH, CH, 1, CH>(&sXq[0][0], &sWt[n0][0], lane);
    store_tile_lds(&sQ[0][n0], UNITS, d, lane);
  }
  // ---- K = Xk(80x64) * Wx^T(64x32) ----
  for (int mt = 0; mt < KEYS / 16; ++mt) {
#pragma unroll
    for (int n0 = 0; n0 < UNITS; n0 += 16) {
      v8f d = wmma_f32_tile<CH, CH, 1, CH>(&sXk[mt * 16][0], &sWx[n0][0], lane);
      store_tile_lds(&sK[mt * 16][n0], UNITS, d, lane);
    }
  }
  __syncthreads();

  // ---- additive scores + banded softmax ----
  // lane L owns query i = L>>1, window half wbase = (L&1)*32
  const float wab   = Wa_b[0];
  const int   qi    = lane >> 1;
  const int   wbase = (lane & 1) * 32;
  const int   s     = s0 + qi;

  float ev[32];
  float m = -__builtin_inff();
  for (int w = 0; w < 32; ++w) {
    const int wr = wbase + w;
    const int t  = s - WIN / 2 + wr;       // key position
    float e;
    if (t >= 0 && t < SEQ) {
      float dot = 0.0f;
#pragma unroll
      for (int u = 0; u < UNITS; ++u)
        dot += sWa[u] * tanhf(sQ[qi][u] + sK[qi + wr][u] + sBh[u]);
      e = dot + wab;
    } else {
      e = -__builtin_inff();
    }
    ev[w] = e;
    m = fmaxf(m, e);
  }
  m = fmaxf(m, __shfl_xor(m, 1, 32));      // join the two half-windows
  float sum = 0.0f;
  for (int w = 0; w < 32; ++w) {
    const float p = (ev[w] == -__builtin_inff()) ? 0.0f : __expf(ev[w] - m);
    ev[w] = p;
    sum += p;
  }
  sum += __shfl_xor(sum, 1, 32);
  const float inv = 1.0f / (sum + EPSF);

  __syncthreads();                          // zeroed sA visible
  for (int w = 0; w < 32; ++w) {
    const int wr = wbase + w;               // band index kidx = qi + wr in [qi, qi+63]
    sA[qi][qi + wr] = ev[w] * inv;
  }
  __syncthreads();

  // ---- write dense `a` rows (mask -> explicit zeros) ----
  float* arow_base = out_a + ((size_t)b * SEQ + s0) * SEQ;
  for (int i2 = 0; i2 < QT; ++i2) {
    float* arow = arow_base + (size_t)i2 * SEQ;
    for (int t = lane; t < SEQ; t += 32) {
      const int kidx = t - t0;
      arow[t] = (kidx >= i2 && kidx < i2 + WIN) ? sA[i2][kidx] : 0.0f;
    }
  }

  // ---- v = A(16x80) * Xk(80x64), store to (B, C, S) layout ----
  const int l  = lane & 15;
  const int mo = (lane >> 4) * 8;
#pragma unroll
  for (int n0 = 0; n0 < CH; n0 += 16) {
    v8f d = wmma_f32_tile<KEYS, KEYS, CH, 1>(&sA[0][0], &sXk[0][0] + n0, lane);
#pragma unroll
    for (int j = 0; j < 8; ++j) {
      const int c  = n0 + l;        // channel
      const int sq = s0 + mo + j;   // sequence position
      out_v[((size_t)b * CH + c) * SEQ + sq] = d[j];
    }
  }
}

extern "C" void kernel_launch(void* const* d_in, const int* in_sizes, int n_in,
                              void* d_out, int out_size, void* d_ws, size_t ws_size,
                              hipStream_t stream) {
  (void)in_sizes; (void)n_in; (void)out_size; (void)d_ws; (void)ws_size;
  const float* x    = (const float*)d_in[0];
  const float* Wt   = (const float*)d_in[1];
  const float* Wx   = (const float*)d_in[2];
  const float* Wa_w = (const float*)d_in[3];
  const float* Wa_b = (const float*)d_in[4];
  const float* bh   = (const float*)d_in[5];

  float* out_v = (float*)d_out;                                   // (B, C, S)
  float* out_a = (float*)d_out + (size_t)BATCH * CH * SEQ;        // (B, S, S)

  dim3 grid(SEQ / QT, BATCH);   // 64 x 4 = 256 workgroups, one wave32 each
  Attention_banded_wmma<<<grid, 32, 0, stream>>>(x, Wt, Wx, Wa_w, Wa_b, bh,
                                                 out_v, out_a);
}